// Clustering_66864050864275
// MI455X (gfx1250) — compile-verified
//
#include <hip/hip_runtime.h>
#include <math.h>

// Problem constants (match reference)
#define B_ 16
#define N_ 2048
#define C_ 1024
#define M_ 128
#define W_ 16   // pool window = N/M
#define EPSN 1e-12f

typedef float v2f __attribute__((ext_vector_type(2)));
typedef float v8f __attribute__((ext_vector_type(8)));

#ifndef __has_builtin
#define __has_builtin(x) 0
#endif
#if __has_builtin(__builtin_amdgcn_global_load_async_to_lds_b128)
#define HAVE_ASYNC_LDS 1
#else
#define HAVE_ASYNC_LDS 0
#endif

#define GLOBAL_AS __attribute__((address_space(1)))
#define LDS_AS __attribute__((address_space(3)))
typedef int b128_t __attribute__((vector_size(16)));  // matches builtin param type

__device__ __forceinline__ void async_cp16(const float* g, float* l) {
#if HAVE_ASYNC_LDS
  // GLOBAL_LOAD_ASYNC_TO_LDS_B128: direct global->LDS, tracked by ASYNCcnt
  __builtin_amdgcn_global_load_async_to_lds_b128(
      (GLOBAL_AS b128_t*)g, (LDS_AS b128_t*)l, 0, 0);
#else
  (void)g; (void)l;
#endif
}

__device__ __forceinline__ void wait_async0() {
#if __has_builtin(__builtin_amdgcn_s_wait_asynccnt)
  __builtin_amdgcn_s_wait_asynccnt(0);
#else
  asm volatile("s_wait_asynccnt 0x0" ::: "memory");
#endif
}

__device__ __forceinline__ float quickgelu(float x) {
  return x / (1.0f + __expf(-1.702f * x));
}

// ---------------------------------------------------------------------------
// fp32 WMMA GEMM:  Out(R x Ncols) = act( opA(A) @ opB(B) + bias )
//   TA=0: A is (R,K) row-major, lda=K.   TA=1: A stored (K,R) row-major, lda=R.
//   TB=0: B is (K,Ncols) row-major, ldb=Ncols. TB=1: B stored (Ncols,K), ldb=K.
// 128x128 tile per 256-thread block (8 waves, each wave: 2x4 tiles of 16x16).
// V_WMMA_F32_16X16X4_F32 (fp32 in/acc) = reference precision on the matrix pipe.
// Double-buffered LDS staging; non-transposed sides use async global->LDS.
// ---------------------------------------------------------------------------
constexpr int APITCH = 20;   // 16 + 4 pad: 16B-aligned, conflict-free A frag reads
constexpr int BPITCH = 144;  // 128 + 16 pad: k / k+2 rows hit disjoint bank groups

template <int TA>
__device__ __forceinline__ void loadA_regs(const float* Ab, int lda, int r0,
                                           int k0, int tid, float4& p0, float4& p1) {
  if (TA == 0) {
    int i = tid >> 1, kb = (tid & 1) * 8;
    const float* s = Ab + (long long)(r0 + i) * lda + k0 + kb;
    p0 = *(const float4*)s; p1 = *(const float4*)(s + 4);
  } else {
    int kk = tid >> 4, ib = (tid & 15) * 8;
    const float* s = Ab + (long long)(k0 + kk) * lda + r0 + ib;
    p0 = *(const float4*)s; p1 = *(const float4*)(s + 4);
  }
}
template <int TA>
__device__ __forceinline__ void storeA_lds(float* As, int tid,
                                           const float4& p0, const float4& p1) {
  if (TA == 0) {
    int i = tid >> 1, kb = (tid & 1) * 8;
    *(float4*)&As[i * APITCH + kb] = p0;
    *(float4*)&As[i * APITCH + kb + 4] = p1;
  } else {
    int kk = tid >> 4, ib = (tid & 15) * 8;
#pragma unroll
    for (int q = 0; q < 4; q++) {
      As[(ib + q) * APITCH + kk] = ((const float*)&p0)[q];
      As[(ib + 4 + q) * APITCH + kk] = ((const float*)&p1)[q];
    }
  }
}
__device__ __forceinline__ void asyncA_issue(const float* Ab, float* AsBuf, int lda,
                                             int r0, int k0, int tid) {
  int i = tid >> 1, kb = (tid & 1) * 8;
  const float* g = Ab + (long long)(r0 + i) * lda + k0 + kb;
  float* l = &AsBuf[i * APITCH + kb];
  async_cp16(g, l);
  async_cp16(g + 4, l + 4);
}

template <int TB>
__device__ __forceinline__ void loadB_regs(const float* Bg, int ldb, int c0,
                                           int k0, int tid, float4& p0, float4& p1) {
  if (TB == 0) {
    int kk = tid >> 5, jb = (tid & 31) * 4;
    p0 = *(const float4*)(Bg + (long long)(k0 + kk) * ldb + c0 + jb);
    p1 = *(const float4*)(Bg + (long long)(k0 + kk + 8) * ldb + c0 + jb);
  } else {
    int j = tid >> 1, kb = (tid & 1) * 8;
    const float* s = Bg + (long long)(c0 + j) * ldb + k0 + kb;
    p0 = *(const float4*)s; p1 = *(const float4*)(s + 4);
  }
}
template <int TB>
__device__ __forceinline__ void storeB_lds(float* Bs, int tid,
                                           const float4& p0, const float4& p1) {
  if (TB == 0) {
    int kk = tid >> 5, jb = (tid & 31) * 4;
    *(float4*)&Bs[kk * BPITCH + jb] = p0;
    *(float4*)&Bs[(kk + 8) * BPITCH + jb] = p1;
  } else {
    int j = tid >> 1, kb = (tid & 1) * 8;
#pragma unroll
    for (int q = 0; q < 4; q++) {
      Bs[(kb + q) * BPITCH + j] = ((const float*)&p0)[q];
      Bs[(kb + 4 + q) * BPITCH + j] = ((const float*)&p1)[q];
    }
  }
}
__device__ __forceinline__ void asyncB_issue(const float* Bg, float* BsBuf, int ldb,
                                             int c0, int k0, int tid) {
  int kk = tid >> 5, jb = (tid & 31) * 4;
  async_cp16(Bg + (long long)(k0 + kk) * ldb + c0 + jb, &BsBuf[kk * BPITCH + jb]);
  async_cp16(Bg + (long long)(k0 + kk + 8) * ldb + c0 + jb, &BsBuf[(kk + 8) * BPITCH + jb]);
}

template <int TA, int TB, int ACT>
__global__ __launch_bounds__(256) void gemm_f32_wmma(
    const float* __restrict__ A, const float* __restrict__ Bm,
    const float* __restrict__ bias, float* __restrict__ Out,
    int K, int lda, int ldb, int ldo,
    long long sA, long long sB, long long sO) {
  __shared__ float As[2][128 * APITCH];
  __shared__ float Bs[2][16 * BPITCH];
  const int tid  = threadIdx.x;
  const int lane = tid & 31, wave = tid >> 5;
  const int wr = (wave >> 1) * 32;   // 4 wave-row groups of 32
  const int wc = (wave & 1) * 64;    // 2 wave-col groups of 64
  const int lrow  = lane & 15;
  const int khalf = lane >> 4;       // which pair of K slices this half-wave holds

  const float* Ab = A  + (long long)blockIdx.z * sA;
  const float* Bg = Bm + (long long)blockIdx.z * sB;
  float*       Ob = Out + (long long)blockIdx.z * sO;
  const int r0 = blockIdx.y * 128, c0 = blockIdx.x * 128;

  const bool asyA = HAVE_ASYNC_LDS && (TA == 0);
  const bool asyB = HAVE_ASYNC_LDS && (TB == 0);
  const bool anyAsync = asyA || asyB;

  v8f acc[2][4];
#pragma unroll
  for (int i = 0; i < 2; i++)
#pragma unroll
    for (int j = 0; j < 4; j++)
#pragma unroll
      for (int v = 0; v < 8; v++) acc[i][j][v] = 0.0f;

  // ---- prologue: fill buffer 0 ----
  {
    float4 a0, a1, b0, b1;
    if (asyA) asyncA_issue(Ab, As[0], lda, r0, 0, tid);
    else      loadA_regs<TA>(Ab, lda, r0, 0, tid, a0, a1);
    if (asyB) asyncB_issue(Bg, Bs[0], ldb, c0, 0, tid);
    else      loadB_regs<TB>(Bg, ldb, c0, 0, tid, b0, b1);
    if (!asyA) storeA_lds<TA>(As[0], tid, a0, a1);
    if (!asyB) storeB_lds<TB>(Bs[0], tid, b0, b1);
    if (anyAsync) wait_async0();
    __syncthreads();
  }

  for (int k0 = 0, buf = 0; k0 < K; k0 += 16, buf ^= 1) {
    const bool pre = (k0 + 16) < K;
    float4 a0, a1, b0, b1;
    // prefetch next stage (async goes straight to LDS; reg path holds in VGPRs)
    if (pre) {
      if (asyA) asyncA_issue(Ab, As[buf ^ 1], lda, r0, k0 + 16, tid);
      else      loadA_regs<TA>(Ab, lda, r0, k0 + 16, tid, a0, a1);
      if (asyB) asyncB_issue(Bg, Bs[buf ^ 1], ldb, c0, k0 + 16, tid);
      else      loadB_regs<TB>(Bg, ldb, c0, k0 + 16, tid, b0, b1);
    }

    // ---- compute on current buffer: 4 K-slices x (2x4) WMMA tiles ----
#pragma unroll
    for (int kk = 0; kk < 16; kk += 4) {
      v2f afrag[2];
#pragma unroll
      for (int mi = 0; mi < 2; mi++)
        afrag[mi] = *(const v2f*)&As[buf][(wr + mi * 16 + lrow) * APITCH + kk + 2 * khalf];
      v2f bfrag[4];
#pragma unroll
      for (int ni = 0; ni < 4; ni++) {
        int col = wc + ni * 16 + lrow;
        bfrag[ni].x = Bs[buf][(kk + 2 * khalf)     * BPITCH + col];
        bfrag[ni].y = Bs[buf][(kk + 2 * khalf + 1) * BPITCH + col];
      }
#pragma unroll
      for (int mi = 0; mi < 2; mi++)
#pragma unroll
        for (int ni = 0; ni < 4; ni++)
          acc[mi][ni] = __builtin_amdgcn_wmma_f32_16x16x4_f32(
              false, afrag[mi], false, bfrag[ni], (short)0, acc[mi][ni],
              false, false);
    }

    if (pre) {
      if (!asyA) storeA_lds<TA>(As[buf ^ 1], tid, a0, a1);
      if (!asyB) storeB_lds<TB>(Bs[buf ^ 1], tid, b0, b1);
      if (anyAsync) wait_async0();
      __syncthreads();
    }
  }

  // epilogue: D layout = VGPR v -> M = v + 8*khalf, N = lane%16
#pragma unroll
  for (int mi = 0; mi < 2; mi++) {
#pragma unroll
    for (int ni = 0; ni < 4; ni++) {
      int col = c0 + wc + ni * 16 + lrow;
      float bcol = bias ? bias[col] : 0.0f;
#pragma unroll
      for (int v = 0; v < 8; v++) {
        int row = r0 + wr + mi * 16 + 8 * khalf + v;
        float val = acc[mi][ni][v] + bcol;
        if (ACT == 1) val = quickgelu(val);
        Ob[(long long)row * ldo + col] = val;
      }
    }
  }
}

// ---------------------------------------------------------------------------
// Adaptive-avg-pool over N (window 16): (B,N,C) -> (B,M,C)
__global__ void pool_kernel(const float* __restrict__ in, float* __restrict__ out) {
  int c = blockIdx.x * 256 + threadIdx.x;
  int m = blockIdx.y, b = blockIdx.z;
  const float* p = in + ((long long)b * N_ + (long long)m * W_) * C_ + c;
  float s = 0.f;
#pragma unroll
  for (int w = 0; w < W_; w++) s += p[(long long)w * C_];
  out[((long long)b * M_ + m) * C_ + c] = s * (1.0f / W_);
}

// colnormN[b][c] = max(||value[b,:,c]||_2, eps)  (folds the over-N normalize)
__global__ void colnorm_kernel(const float* __restrict__ v, float* __restrict__ nrm) {
  int c = blockIdx.x * 256 + threadIdx.x;
  int b = blockIdx.y;
  const float* p = v + (long long)b * N_ * C_ + c;
  float ss = 0.f;
  for (int n = 0; n < N_; n++) { float t = p[(long long)n * C_]; ss += t * t; }
  nrm[b * C_ + c] = fmaxf(sqrtf(ss), EPSN);
}

// Row L2-normalize (optionally also divide by a per-(batch,c) column scale),
// optionally only emit the row norms.
__global__ __launch_bounds__(256) void rownorm_kernel(
    const float* __restrict__ in, const float* __restrict__ colscale,
    float* __restrict__ outv, float* __restrict__ outn, int len, int rowsPerBatch) {
  __shared__ float red[256];
  long long row = blockIdx.x;
  const float* p = in + row * len;
  int tid = threadIdx.x;
  float ss = 0.f;
  for (int c = tid; c < len; c += 256) { float t = p[c]; ss += t * t; }
  red[tid] = ss; __syncthreads();
  for (int off = 128; off > 0; off >>= 1) {
    if (tid < off) red[tid] += red[tid + off];
    __syncthreads();
  }
  float nrm = fmaxf(sqrtf(red[0]), EPSN);
  if (outn && tid == 0) outn[row] = nrm;
  if (outv) {
    float inv = 1.0f / nrm;
    const float* cs = colscale ? colscale + (row / rowsPerBatch) * len : nullptr;
    for (int c = tid; c < len; c += 256)
      outv[row * len + c] = cs ? p[c] * inv / cs[c] : p[c] * inv;
  }
}

// Softmax over the contiguous M=128 dim of sim stored (B,N,M)
__global__ __launch_bounds__(128) void softmax128_kernel(float* __restrict__ s) {
  __shared__ float red[128];
  long long row = blockIdx.x; int tid = threadIdx.x;
  float v = s[row * 128 + tid];
  red[tid] = v; __syncthreads();
  for (int off = 64; off > 0; off >>= 1) {
    if (tid < off) red[tid] = fmaxf(red[tid], red[tid + off]);
    __syncthreads();
  }
  float mx = red[0]; __syncthreads();
  float e = __expf(v - mx);
  red[tid] = e; __syncthreads();
  for (int off = 64; off > 0; off >>= 1) {
    if (tid < off) red[tid] += red[tid + off];
    __syncthreads();
  }
  s[row * 128 + tid] = e / red[0];
}

// sim holds cos-dot (B,N,M); apply /||x_row||, sigmoid(beta+alpha*..),
// first-max argmax over M, rewrite row in place as the hard-masked similarity.
__global__ __launch_bounds__(128) void sig_argmax_kernel(
    float* __restrict__ sim, const float* __restrict__ xnorm,
    const float* __restrict__ alpha, const float* __restrict__ beta,
    int* __restrict__ idxout) {
  __shared__ float vals[128];
  __shared__ int   idxs[128];
  long long row = blockIdx.x; int tid = threadIdx.x;
  float a = alpha[0], bt = beta[0];
  float inv = 1.0f / xnorm[row];
  float s = 1.0f / (1.0f + __expf(-(bt + a * sim[row * 128 + tid] * inv)));
  vals[tid] = s; idxs[tid] = tid; __syncthreads();
  for (int off = 64; off > 0; off >>= 1) {
    if (tid < off) {
      float vo = vals[tid + off]; int io = idxs[tid + off];
      if (vo > vals[tid] || (vo == vals[tid] && io < idxs[tid])) {
        vals[tid] = vo; idxs[tid] = io;   // first-max semantics (torch.max)
      }
    }
    __syncthreads();
  }
  int best = idxs[0];
  __syncthreads();
  sim[row * 128 + tid] = (tid == best) ? s : 0.0f;
  if (tid == 0) idxout[row] = best;
}

// counts[b][m] = #{n : idx[b][n]==m}  (integer LDS atomics: deterministic)
__global__ __launch_bounds__(256) void counts_kernel(const int* __restrict__ idx,
                                                     int* __restrict__ counts) {
  __shared__ int cnt[M_];
  int tid = threadIdx.x, b = blockIdx.x;
  if (tid < M_) cnt[tid] = 0;
  __syncthreads();
  for (int n = tid; n < N_; n += 256) atomicAdd(&cnt[idx[(long long)b * N_ + n]], 1);
  __syncthreads();
  if (tid < M_) counts[b * M_ + tid] = cnt[tid];
}

// agg = (agg + centers_feature) / (counts + 1)
__global__ void agg_finalize_kernel(float* __restrict__ agg,
                                    const float* __restrict__ cf,
                                    const int* __restrict__ counts) {
  int c = blockIdx.x * 256 + threadIdx.x;
  int m = blockIdx.y, b = blockIdx.z;
  long long i = ((long long)b * M_ + m) * C_ + c;
  float inv = 1.0f / (float)(counts[b * M_ + m] + 1);
  agg[i] = (agg[i] + cf[i]) * inv;
}

// out[b,n,:] = x[b,n,:] + agg[b,idx[b,n],:] * smask[b,n,idx]
__global__ __launch_bounds__(256) void scatter_out_kernel(
    const float* __restrict__ x, const float* __restrict__ agg,
    const int* __restrict__ idx, const float* __restrict__ smask,
    float* __restrict__ out) {
  long long row = blockIdx.x;           // b*N + n
  int b = (int)(row >> 11);             // N_ == 2048
  int m = idx[row];
  float s = smask[row * 128 + m];
  int tid = threadIdx.x;                // 256 thr * float4 = 1024 = C_
  float4 xv = ((const float4*)(x + row * C_))[tid];
  float4 av = ((const float4*)(agg + ((long long)b * M_ + m) * C_))[tid];
  float4 r;
  r.x = xv.x + av.x * s; r.y = xv.y + av.y * s;
  r.z = xv.z + av.z * s; r.w = xv.w + av.w * s;
  ((float4*)(out + row * C_))[tid] = r;
}

// ---------------------------------------------------------------------------
extern "C" void kernel_launch(void* const* d_in, const int* in_sizes, int n_in,
                              void* d_out, int out_size, void* d_ws, size_t ws_size,
                              hipStream_t stream) {
  (void)in_sizes; (void)n_in; (void)out_size; (void)ws_size;
  const float* x     = (const float*)d_in[0];
  const float* Wv    = (const float*)d_in[1];
  const float* bv    = (const float*)d_in[2];
  const float* Wf    = (const float*)d_in[3];
  const float* bf    = (const float*)d_in[4];
  const float* Wc    = (const float*)d_in[5];
  const float* bc    = (const float*)d_in[6];
  const float* W1    = (const float*)d_in[7];
  const float* b1    = (const float*)d_in[8];
  const float* W2    = (const float*)d_in[9];
  const float* b2    = (const float*)d_in[10];
  const float* alpha = (const float*)d_in[11];
  const float* beta  = (const float*)d_in[12];
  float* y = (float*)d_out;

  // workspace bump allocator (256B aligned)
  char* wsp = (char*)d_ws;
  auto alloc = [&](size_t bytes) -> void* {
    void* p = (void*)wsp;
    wsp += (bytes + 255) & ~(size_t)255;
    return p;
  };
  const long long BNC = (long long)B_ * N_ * C_;
  const long long BMC = (long long)B_ * M_ * C_;
  const long long BNM = (long long)B_ * N_ * M_;
  float* F1       = (float*)alloc(BNC * 4);  // value -> (reuse) out
  float* F2       = (float*)alloc(BNC * 4);  // feature -> (reuse) h
  float* sim      = (float*)alloc(BNM * 4);  // St / cos / smask (B,N,M)
  float* centersA = (float*)alloc(BMC * 4);
  float* centersB = (float*)alloc(BMC * 4);
  float* cn       = (float*)alloc(BMC * 4);  // cn' / chat
  float* cf       = (float*)alloc(BMC * 4);  // centers_feature
  float* agg      = (float*)alloc(BMC * 4);
  float* colnormN = (float*)alloc((long long)B_ * C_ * 4);
  float* xnorm    = (float*)alloc((long long)B_ * N_ * 4);
  int*   idx      = (int*)alloc((long long)B_ * N_ * 4);
  int*   counts   = (int*)alloc((long long)B_ * M_ * 4);

  const dim3 blk256(256), blk128(128);
  const dim3 gBig(C_ / 128, (B_ * N_) / 128, 1);      // 32768-row weight GEMMs
  const dim3 gCen(C_ / 128, (B_ * M_) / 128, 1);      // 2048-row centers GEMM
  const dim3 gNT(M_ / 128, N_ / 128, B_);             // sim / cos NT GEMMs
  const dim3 gTN(C_ / 128, M_ / 128, B_);             // simT @ feature TN GEMMs

  // 1) value = x@Wv+bv ; feature = x@Wf+bf   (fp32 WMMA)
  gemm_f32_wmma<0, 0, 0><<<gBig, blk256, 0, stream>>>(x, Wv, bv, F1, C_, C_, C_, C_, 0, 0, 0);
  gemm_f32_wmma<0, 0, 0><<<gBig, blk256, 0, stream>>>(x, Wf, bf, F2, C_, C_, C_, C_, 0, 0, 0);

  // 2) pools
  pool_kernel<<<dim3(C_ / 256, M_, B_), blk256, 0, stream>>>(x, centersA);
  pool_kernel<<<dim3(C_ / 256, M_, B_), blk256, 0, stream>>>(F2, cf);

  // 3) per-(b,c) norm of value over N (normalize along N folded into cn')
  colnorm_kernel<<<dim3(C_ / 256, B_), blk256, 0, stream>>>(F1, colnormN);

  // 4) refinement iterations
  for (int it = 0; it < 3; it++) {
    gemm_f32_wmma<0, 0, 0><<<gCen, blk256, 0, stream>>>(centersA, Wc, bc, centersB,
                                                        C_, C_, C_, C_, 0, 0, 0);
    // cn' = rownorm(centers)/colnormN  -> sim = value @ cn'^T uses RAW value
    rownorm_kernel<<<dim3(B_ * M_), blk256, 0, stream>>>(centersB, colnormN, cn,
                                                         nullptr, C_, M_);
    gemm_f32_wmma<0, 1, 0><<<gNT, blk256, 0, stream>>>(F1, cn, nullptr, sim,
        C_, C_, C_, M_, (long long)N_ * C_, (long long)M_ * C_, (long long)N_ * M_);
    softmax128_kernel<<<dim3(B_ * N_), blk128, 0, stream>>>(sim);
    // centers = sim^T @ feature  (TN)
    gemm_f32_wmma<1, 0, 0><<<gTN, blk256, 0, stream>>>(sim, F2, nullptr, centersA,
        N_, M_, C_, C_, (long long)N_ * M_, (long long)N_ * C_, (long long)M_ * C_);
  }

  // 5) chat = rownorm(centers); xnorm rows; cos = x @ chat^T  (x-norm folded later)
  rownorm_kernel<<<dim3(B_ * M_), blk256, 0, stream>>>(centersA, nullptr, cn, nullptr, C_, M_);
  rownorm_kernel<<<dim3(B_ * N_), blk256, 0, stream>>>(x, nullptr, nullptr, xnorm, C_, N_);
  gemm_f32_wmma<0, 1, 0><<<gNT, blk256, 0, stream>>>(x, cn, nullptr, sim,
      C_, C_, C_, M_, (long long)N_ * C_, (long long)M_ * C_, (long long)N_ * M_);

  // 6) sigmoid + hard top-1 (in place -> sparse masked similarity), counts
  sig_argmax_kernel<<<dim3(B_ * N_), blk128, 0, stream>>>(sim, xnorm, alpha, beta, idx);
  counts_kernel<<<dim3(B_), blk256, 0, stream>>>(idx, counts);

  // 7) agg = smask^T @ feature ; (agg+cf)/(counts+1) ; out = x + gather
  gemm_f32_wmma<1, 0, 0><<<gTN, blk256, 0, stream>>>(sim, F2, nullptr, agg,
      N_, M_, C_, C_, (long long)N_ * M_, (long long)N_ * C_, (long long)M_ * C_);
  agg_finalize_kernel<<<dim3(C_ / 256, M_, B_), blk256, 0, stream>>>(agg, cf, counts);
  scatter_out_kernel<<<dim3(B_ * N_), blk256, 0, stream>>>(x, agg, idx, sim, F1);

  // 8) MLP: h = QuickGELU(out@W1+b1) ; y = h@W2+b2
  gemm_f32_wmma<0, 0, 1><<<gBig, blk256, 0, stream>>>(F1, W1, b1, F2, C_, C_, C_, C_, 0, 0, 0);
  gemm_f32_wmma<0, 0, 0><<<gBig, blk256, 0, stream>>>(F2, W2, b2, y, C_, C_, C_, C_, 0, 0, 0);
}